// SphericalModelDeep_43817256354397
// MI455X (gfx1250) — compile-verified
//
// MI455X / gfx1250 implementation of the 14-layer spherical CNN reference.
//
// Strategy (reasoned from MI455X specs, compile-only):
//  * The hot op per layer is a per-degree-l complex GEMM:
//      yh[b,o,l,m,n] = sum_{i,k} xh[b,i,l,m,k] * conj(psi)[o,i,l,n,k]
//    Mapped to V_WMMA_F32_16X16X4_F32 (fp32 A/B, fp32 accum) to preserve the
//    reference's complex64 numerics: 4 real WMMAs per complex K=4 step.
//  * GEMM inner loop is division-free: channel-major outer loop with linear
//    K addressing, 2x2 tile (32x32 C) per wave32 -> 16 WMMAs per K-step and
//    2x reuse of every fragment. K-tail masking is hoisted out of the hot
//    loop; row/col edge handling is clamp-only (garbage lands in C rows/cols
//    that are never stored), so EXEC stays all-1s at every WMMA.
//  * Everything else (DFT passes as twiddle-table matmuls, Wigner-d synthesis,
//    bias+ReLU, pooling, BN/MLP head) is VALU work; data is L2-resident
//    (192 MB), so HBM (23.3 TB/s) is never the limiter -> fp32 throughout.
//  * All module-level constants (quad weights, Wigner-d via log-factorial
//    formula, kernel bases, twiddles) are regenerated on-device each call:
//    deterministic, no host I/O, graph-capture safe.

#include <hip/hip_runtime.h>
#include <math.h>

#ifndef M_PI
#define M_PI 3.14159265358979323846
#endif

typedef float v2f __attribute__((ext_vector_type(2)));
typedef float v8f __attribute__((ext_vector_type(8)));

#define GSL(i, n) for (long i = blockIdx.x * (long)blockDim.x + threadIdx.x; \
                       i < (n); i += (long)gridDim.x * blockDim.x)

__host__ __device__ static inline int imin_(int a, int b) { return a < b ? a : b; }
__host__ __device__ static inline int imax_(int a, int b) { return a > b ? a : b; }

// ---------------------------------------------------------------------------
// Constant generation
// ---------------------------------------------------------------------------

// ln(n!) table, n in [0,127]
__global__ void k_lgfact(double* lgf) {
  int n = blockIdx.x * blockDim.x + threadIdx.x;
  if (n < 128) lgf[n] = lgamma((double)n + 1.0);
}

// Wigner small-d via the explicit (log-factorial) sum; stable for l<=29.
__device__ static double wig_d(int l, int mp, int m, double beta, const double* lgf) {
  int kmin = imax_(0, m - mp);
  int kmax = imin_(l + m, l - mp);
  if (kmax < kmin) return 0.0;
  double cb = cos(0.5 * beta), sb = sin(0.5 * beta);
  double pref = 0.5 * (lgf[l + mp] + lgf[l - mp] + lgf[l + m] + lgf[l - m]);
  double acc = 0.0;
  for (int k = kmin; k <= kmax; ++k) {
    double ln = pref - lgf[l + m - k] - lgf[k] - lgf[mp - m + k] - lgf[l - mp - k];
    double t = exp(ln) * pow(cb, (double)(2 * l + m - mp - 2 * k))
                       * pow(sb, (double)(mp - m + 2 * k));
    acc += ((mp - m + k) & 1) ? -t : t;
  }
  return acc;
}

// Driscoll-Healy quadrature weights, w[k] for k in [0,2b)
__global__ void k_quadw(float* w, int b) {
  int k = blockIdx.x * blockDim.x + threadIdx.x;
  if (k >= 2 * b) return;
  double beta = M_PI * (2 * k + 1) / (4.0 * b);
  double s = 0.0;
  for (int j = 0; j < b; ++j) s += sin(beta * (2 * j + 1)) / (2 * j + 1);
  w[k] = (float)(2.0 / b * sin(beta) * s);
}

// Padded SO3 Wigner table: out[l, m, n, k]  (M = 2nl-1, k < 2b)
// useW=1: Df  (x quad weight of b);  useW=0: Di  (x (l+0.5))
__global__ void k_tab_so3(float* out, const float* wq, int b, int nl, int useW,
                          const double* lgf) {
  int M = 2 * nl - 1, K2 = 2 * b;
  long tot = (long)nl * M * M * K2;
  GSL(i, tot) {
    int k = (int)(i % K2); long r = i / K2;
    int n = (int)(r % M);  r /= M;
    int m = (int)(r % M);  int l = (int)(r / M);
    int c = nl - 1, a = m - c, g = n - c;
    float v = 0.f;
    if (a >= -l && a <= l && g >= -l && g <= l) {
      double beta = M_PI * (2 * k + 1) / (4.0 * b);
      double d = wig_d(l, a, g, beta, lgf);
      v = (float)(useW ? d * (double)wq[k] : d * ((double)l + 0.5));
    }
    out[i] = v;
  }
}

// Padded S2 Df table: out[l, m, k] = d^l_{m-c,0}(beta_k) * w[k]
__global__ void k_tab_s2df(float* out, const float* wq, int b, int nl,
                           const double* lgf) {
  int M = 2 * nl - 1, K2 = 2 * b;
  long tot = (long)nl * M * K2;
  GSL(i, tot) {
    int k = (int)(i % K2); long r = i / K2;
    int m = (int)(r % M);  int l = (int)(r / M);
    int a = m - (nl - 1);
    float v = 0.f;
    if (a >= -l && a <= l) {
      double beta = M_PI * (2 * k + 1) / (4.0 * b);
      v = (float)(wig_d(l, a, 0, beta, lgf) * (double)wq[k]);
    }
    out[i] = v;
  }
}

// SO3 kernel basis on the 36-point grid: K[l,m,n,p] = d * e^{-i(a*alpha+g*gamma)}
__global__ void k_kb_so3(float* Kr, float* Ki, int nl, double gbeta,
                         const double* lgf) {
  int M = 2 * nl - 1; const int P = 36;
  long tot = (long)nl * M * M * P;
  GSL(i, tot) {
    int p = (int)(i % P); long r = i / P;
    int n = (int)(r % M); r /= M;
    int m = (int)(r % M); int l = (int)(r / M);
    int c = nl - 1, a = m - c, g = n - c;
    float vr = 0.f, vi = 0.f;
    if (a >= -l && a <= l && g >= -l && g <= l) {
      double alpha = 2.0 * M_PI * (double)(p / 6) / 6.0;
      double gamma = 2.0 * M_PI * (double)(p % 6) / 6.0;
      double d  = wig_d(l, a, g, gbeta, lgf);
      double ph = -((double)a * alpha + (double)g * gamma);
      vr = (float)(d * cos(ph)); vi = (float)(d * sin(ph));
    }
    Kr[i] = vr; Ki[i] = vi;
  }
}

// S2 kernel basis on 6-point grid: K[l,m,p] = d^l_{a,0}(beta) e^{-i a alpha_p}
__global__ void k_kb_s2(float* Kr, float* Ki, int nl, double gbeta,
                        const double* lgf) {
  int M = 2 * nl - 1; const int P = 6;
  long tot = (long)nl * M * P;
  GSL(i, tot) {
    int p = (int)(i % P); long r = i / P;
    int m = (int)(r % M); int l = (int)(r / M);
    int a = m - (nl - 1);
    float vr = 0.f, vi = 0.f;
    if (a >= -l && a <= l) {
      double alpha = 2.0 * M_PI * (double)p / 6.0;
      double d = wig_d(l, a, 0, gbeta, lgf);
      double ph = -(double)a * alpha;
      vr = (float)(d * cos(ph)); vi = (float)(d * sin(ph));
    }
    Kr[i] = vr; Ki[i] = vi;
  }
}

// Twiddle table tw[f, g] = e^{sgn * 2pi i * (f-(nl-1)) * g / K2}
__global__ void k_twiddle(float* twr, float* twi, int M, int K2, int nl, double sgn) {
  long tot = (long)M * K2;
  GSL(i, tot) {
    int g = (int)(i % K2); int m = (int)(i / K2);
    double ang = sgn * 2.0 * M_PI * (double)(m - (nl - 1)) * (double)g / (double)K2;
    twr[i] = (float)cos(ang); twi[i] = (float)sin(ang);
  }
}

// ---------------------------------------------------------------------------
// Transform kernels
// ---------------------------------------------------------------------------

// out[ld, m] = scale * sum_g in[ld, g] * tw[m, g]    (real input DFT, last axis)
__global__ void k_dft_real(float* outR, float* outI, const float* in,
                           const float* twr, const float* twi,
                           long lead, int K2, int M, float scale) {
  long tot = lead * M;
  GSL(i, tot) {
    int m = (int)(i % M); long ld = i / M;
    const float* ip = in + ld * K2;
    const float* tr = twr + (long)m * K2;
    const float* ti = twi + (long)m * K2;
    float sr = 0.f, si = 0.f;
    for (int g = 0; g < K2; ++g) { float xv = ip[g]; sr += xv * tr[g]; si += xv * ti[g]; }
    outR[i] = sr * scale; outI[i] = si * scale;
  }
}

// out[q, m, n] = scale * sum_a tw[m, a] * in[q, a, n]  (complex, middle axis)
__global__ void k_dft_mid(float* outR, float* outI,
                          const float* inR, const float* inI,
                          const float* twr, const float* twi,
                          long L1, int A, int N, int Mo, float scale) {
  long tot = L1 * (long)Mo * N;
  GSL(i, tot) {
    int n = (int)(i % N); long r = i / N;
    int m = (int)(r % Mo); long q = r / Mo;
    const float* tr = twr + (long)m * A;
    const float* ti = twi + (long)m * A;
    float sr = 0.f, si = 0.f;
    for (int a = 0; a < A; ++a) {
      long xi = (q * A + a) * (long)N + n;
      float xr = inR[xi], xim = inI[xi];
      sr += tr[a] * xr - ti[a] * xim;
      si += tr[a] * xim + ti[a] * xr;
    }
    outR[i] = sr * scale; outI[i] = si * scale;
  }
}

// xh[bc, l, s] = sum_k Df[l, s, k] * Xs[bc, k, s]   (s spans SP2 spatial elems)
__global__ void k_xh(float* xhr, float* xhi,
                     const float* Xsr, const float* Xsi,
                     const float* Df, long BC, int nl, long SP2, int K2) {
  long tot = BC * nl * SP2;
  GSL(i, tot) {
    long r2 = i % SP2; long r = i / SP2;
    int l = (int)(r % nl); long bc = r / nl;
    const float* dp = Df + ((long)l * SP2 + r2) * K2;
    float sr = 0.f, si = 0.f;
    for (int k = 0; k < K2; ++k) {
      float d = dp[k];
      long xi = (bc * K2 + k) * SP2 + r2;
      sr += d * Xsr[xi]; si += d * Xsi[xi];
    }
    xhr[i] = sr; xhi[i] = si;
  }
}

// psiC[oi, s] = conj( sum_p w[oi, p] * Kb[s, p] )
__global__ void k_psi(float* pr, float* pim, const float* w,
                      const float* Kr, const float* Ki, long OI, long SP, int P) {
  long tot = OI * SP;
  GSL(i, tot) {
    long s = i % SP; long oi = i / SP;
    const float* wp = w + oi * P;
    const float* kr = Kr + s * P;
    const float* ki = Ki + s * P;
    float sr = 0.f, si = 0.f;
    for (int p = 0; p < P; ++p) { float ww = wp[p]; sr += ww * kr[p]; si += ww * ki[p]; }
    pr[i] = sr; pim[i] = -si;   // store conj(psi)
  }
}

// ---------------------------------------------------------------------------
// WMMA complex GEMM (per-degree l batched):
//   C[(b,m),(o,n)] = sum_{i,k} A[b,i,l,m,k] * B[o,i,l,n,k]   (complex)
// A = xh (B,CI,nl,M,Kext), B = conj(psi) (CO,CI,nl,M,Kext), C = yh (B,CO,nl,M,M)
// One wave32 computes a 32x32 C tile (2x2 WMMA tiles) with fp32 WMMA.
// ---------------------------------------------------------------------------

__device__ __forceinline__ void cwmma(v8f& cre, v8f& cim, v2f ar, v2f ai,
                                      v2f br, v2f bi) {
  v2f nai; nai[0] = -ai[0]; nai[1] = -ai[1];
  // Cr += Ar*Br - Ai*Bi ; Ci += Ar*Bi + Ai*Br
  cre = __builtin_amdgcn_wmma_f32_16x16x4_f32(false, ar,  false, br, (short)0, cre, false, false);
  cre = __builtin_amdgcn_wmma_f32_16x16x4_f32(false, nai, false, bi, (short)0, cre, false, false);
  cim = __builtin_amdgcn_wmma_f32_16x16x4_f32(false, ar,  false, bi, (short)0, cim, false, false);
  cim = __builtin_amdgcn_wmma_f32_16x16x4_f32(false, ai,  false, br, (short)0, cim, false, false);
}

__device__ __forceinline__ v2f ld2(const float* p, int k0) {
  v2f r; r[0] = p[k0]; r[1] = p[k0 + 1]; return r;
}

__device__ __forceinline__ void cstore(float* __restrict__ Cr, float* __restrict__ Ci,
                                       int row, int col, int Rtot, int Stot,
                                       int CO, int nl, int l, int M,
                                       float vr, float vi) {
  if (row < Rtot && col < Stot) {
    int cb = row / M, cm = row % M;
    int co = col / M, cn = col % M;
    long cidx = ((((long)cb * CO + co) * nl + l) * M + cm) * (long)M + cn;
    Cr[cidx] = vr; Ci[cidx] = vi;
  }
}

__global__ __launch_bounds__(32) void k_cgemm_wmma(
    const float* __restrict__ Ar, const float* __restrict__ Ai,
    const float* __restrict__ Br, const float* __restrict__ Bi,
    float* __restrict__ Cr, float* __restrict__ Ci,
    int Bn, int CI, int CO, int nl, int M, int Kext)
{
  const int l    = blockIdx.z;
  const int lane = threadIdx.x;
  const int rt   = blockIdx.y * 32;          // 2 row tiles of 16
  const int st   = blockIdx.x * 32;          // 2 col tiles of 16
  const int Rtot = Bn * M, Stot = CO * M;

  const int rloc  = lane & 15;
  const int half2 = (lane >> 4) * 2;         // A/B fragment K sub-offset

  // Clamp-only edge handling: an out-of-range row/col reads in-bounds data
  // and produces garbage only in C rows/cols that are never stored below.
  const int ar0 = imin_(rt + rloc,      Rtot - 1);
  const int ar1 = imin_(rt + 16 + rloc, Rtot - 1);
  const int bc0 = imin_(st + rloc,      Stot - 1);
  const int bc1 = imin_(st + 16 + rloc, Stot - 1);

  const long sCh = (long)nl * M * Kext;      // channel stride (A and B)
  const long sL  = (long)M * Kext;           // degree stride within channel

  const long aBase0 = (long)(ar0 / M) * CI * sCh + (long)l * sL + (long)(ar0 % M) * Kext;
  const long aBase1 = (long)(ar1 / M) * CI * sCh + (long)l * sL + (long)(ar1 % M) * Kext;
  const long bBase0 = (long)(bc0 / M) * CI * sCh + (long)l * sL + (long)(bc0 % M) * Kext;
  const long bBase1 = (long)(bc1 / M) * CI * sCh + (long)l * sL + (long)(bc1 % M) * Kext;

  v8f cr00 = {}, ci00 = {}, cr01 = {}, ci01 = {};
  v8f cr10 = {}, ci10 = {}, cr11 = {}, ci11 = {};

  for (int i = 0; i < CI; ++i) {
    const long chOff = (long)i * sCh;
    const float* a0r = Ar + aBase0 + chOff + half2;
    const float* a0i = Ai + aBase0 + chOff + half2;
    const float* a1r = Ar + aBase1 + chOff + half2;
    const float* a1i = Ai + aBase1 + chOff + half2;
    const float* b0r = Br + bBase0 + chOff + half2;
    const float* b0i = Bi + bBase0 + chOff + half2;
    const float* b1r = Br + bBase1 + chOff + half2;
    const float* b1i = Bi + bBase1 + chOff + half2;

    int k0 = 0;
    // Hot loop: linear addressing, no masking, 16 WMMAs per step.
    for (; k0 + 4 <= Kext; k0 += 4) {
      if (k0 + 8 <= Kext) {                  // global_prefetch_b8 next slab
        __builtin_prefetch(a0r + k0 + 4, 0, 1);
        __builtin_prefetch(b0r + k0 + 4, 0, 1);
      }
      v2f A0r = ld2(a0r, k0), A0i = ld2(a0i, k0);
      v2f A1r = ld2(a1r, k0), A1i = ld2(a1i, k0);
      v2f B0r = ld2(b0r, k0), B0i = ld2(b0i, k0);
      v2f B1r = ld2(b1r, k0), B1i = ld2(b1i, k0);
      cwmma(cr00, ci00, A0r, A0i, B0r, B0i);
      cwmma(cr01, ci01, A0r, A0i, B1r, B1i);
      cwmma(cr10, ci10, A1r, A1i, B0r, B0i);
      cwmma(cr11, ci11, A1r, A1i, B1r, B1i);
    }
    // Masked K-tail (Kext % 4 != 0, incl. the S2 layer's Kext == 1).
    if (k0 < Kext) {
      v2f A0r, A0i, A1r, A1i, B0r, B0i, B1r, B1i;
#pragma unroll
      for (int v = 0; v < 2; ++v) {
        int kk = k0 + half2 + v;
        float f = (kk < Kext) ? 1.f : 0.f;
        int kc = (kk < Kext) ? kk : 0;
        A0r[v] = Ar[aBase0 + chOff + kc] * f;
        A0i[v] = Ai[aBase0 + chOff + kc] * f;
        A1r[v] = Ar[aBase1 + chOff + kc] * f;
        A1i[v] = Ai[aBase1 + chOff + kc] * f;
        B0r[v] = Br[bBase0 + chOff + kc] * f;
        B0i[v] = Bi[bBase0 + chOff + kc] * f;
        B1r[v] = Br[bBase1 + chOff + kc] * f;
        B1i[v] = Bi[bBase1 + chOff + kc] * f;
      }
      cwmma(cr00, ci00, A0r, A0i, B0r, B0i);
      cwmma(cr01, ci01, A0r, A0i, B1r, B1i);
      cwmma(cr10, ci10, A1r, A1i, B0r, B0i);
      cwmma(cr11, ci11, A1r, A1i, B1r, B1i);
    }
  }

  // C/D layout: VGPR r, lanes 0-15 -> row base+r, lanes 16-31 -> row base+r+8
  const int half8 = (lane >> 4) * 8;
#pragma unroll
  for (int r = 0; r < 8; ++r) {
    int rowa = rt + r + half8;
    int rowb = rt + 16 + r + half8;
    int col0 = st + rloc;
    int col1 = st + 16 + rloc;
    cstore(Cr, Ci, rowa, col0, Rtot, Stot, CO, nl, l, M, cr00[r], ci00[r]);
    cstore(Cr, Ci, rowa, col1, Rtot, Stot, CO, nl, l, M, cr01[r], ci01[r]);
    cstore(Cr, Ci, rowb, col0, Rtot, Stot, CO, nl, l, M, cr10[r], ci10[r]);
    cstore(Cr, Ci, rowb, col1, Rtot, Stot, CO, nl, l, M, cr11[r], ci11[r]);
  }
}

// ---------------------------------------------------------------------------
// Inverse path
// ---------------------------------------------------------------------------

// Ys[bo, kq, s] = sum_l Di[l, s, kq] * yh[bo, l, s]
__global__ void k_ys(float* Ysr, float* Ysi,
                     const float* yhr, const float* yhi,
                     const float* Di, long BO, int nl, long SP2, int K2o) {
  long tot = BO * K2o * SP2;
  GSL(i, tot) {
    long r2 = i % SP2; long r = i / SP2;
    int kq = (int)(r % K2o); long bo = r / K2o;
    float sr = 0.f, si = 0.f;
    for (int l = 0; l < nl; ++l) {
      float d = Di[((long)l * SP2 + r2) * K2o + kq];
      long yi = (bo * nl + l) * SP2 + r2;
      sr += d * yhr[yi]; si += d * yhi[yi];
    }
    Ysr[i] = sr; Ysi[i] = si;
  }
}

// out[ld, g] = sum_n in[ld, n] * tw[n, g]   (complex, last axis, no scale)
__global__ void k_idft_last(float* outR, float* outI,
                            const float* inR, const float* inI,
                            const float* twr, const float* twi,
                            long lead, int M, int K2o) {
  long tot = lead * K2o;
  GSL(i, tot) {
    int g = (int)(i % K2o); long ld = i / K2o;
    float sr = 0.f, si = 0.f;
    for (int n = 0; n < M; ++n) {
      float xr = inR[ld * M + n], xim = inI[ld * M + n];
      float tr = twr[(long)n * K2o + g], ti = twi[(long)n * K2o + g];
      sr += xr * tr - xim * ti;
      si += xr * ti + xim * tr;
    }
    outR[i] = sr; outI[i] = si;
  }
}

// y[p, a, g] = relu( Re( sum_m tw[m, a] * U[p, m, g] ) + bias[o] )
__global__ void k_idft_mid_bias_relu(float* y, const float* Ur, const float* Ui,
                                     const float* twr, const float* twi,
                                     const float* bias, long Prows, int M,
                                     int K2o, int CO) {
  long tot = Prows * (long)K2o * K2o;
  GSL(i, tot) {
    int g = (int)(i % K2o); long r = i / K2o;
    int a = (int)(r % K2o); long p = r / K2o;
    int o = (int)((p / K2o) % CO);
    float s = 0.f;
    for (int m = 0; m < M; ++m) {
      float tr = twr[(long)m * K2o + a], ti = twi[(long)m * K2o + a];
      long u = (p * M + m) * (long)K2o + g;
      s += tr * Ur[u] - ti * Ui[u];
    }
    s += bias[o];
    y[i] = fmaxf(s, 0.f);
  }
}

// feat[bc] = sum_{k,m,n} h[bc, k, m, n] * w30[k]   (one block per (b,c))
__global__ void k_feat(float* feat, const float* h, const float* w30, int K2) {
  int bc = blockIdx.x;
  long n = (long)K2 * K2 * K2;
  const float* hp = h + (long)bc * n;
  float s = 0.f;
  long kk2 = (long)K2 * K2;
  for (long i = threadIdx.x; i < n; i += blockDim.x) {
    int kq = (int)(i / kk2);
    s += hp[i] * w30[kq];
  }
  __shared__ float red[256];
  red[threadIdx.x] = s; __syncthreads();
  for (int d = 128; d > 0; d >>= 1) {
    if ((int)threadIdx.x < d) red[threadIdx.x] += red[threadIdx.x + d];
    __syncthreads();
  }
  if (threadIdx.x == 0) feat[bc] = red[0];
}

// BN -> lin(64) -> relu -> BN -> lin(32) -> relu -> BN -> lin(10); B=2
__global__ void k_head(const float* feat,
                       const float* g1, const float* be1, const float* lw1, const float* lb1,
                       const float* g2, const float* be2, const float* lw2, const float* lb2,
                       const float* g3, const float* be3, const float* lw3, const float* lb3,
                       float* out) {
  __shared__ float z0[2][64], z1[2][64], z2[2][32];
  int t = threadIdx.x;
  {
    float f0 = feat[t], f1 = feat[64 + t];
    float mu = 0.5f * (f0 + f1);
    float var = 0.5f * ((f0 - mu) * (f0 - mu) + (f1 - mu) * (f1 - mu));
    float is = rsqrtf(var + 1e-5f);
    z0[0][t] = g1[t] * (f0 - mu) * is + be1[t];
    z0[1][t] = g1[t] * (f1 - mu) * is + be1[t];
  }
  __syncthreads();
  for (int b = 0; b < 2; ++b) {
    float s = lb1[t];
    for (int k = 0; k < 64; ++k) s += z0[b][k] * lw1[t * 64 + k];
    z1[b][t] = fmaxf(s, 0.f);
  }
  __syncthreads();
  {
    float f0 = z1[0][t], f1 = z1[1][t];
    float mu = 0.5f * (f0 + f1);
    float var = 0.5f * ((f0 - mu) * (f0 - mu) + (f1 - mu) * (f1 - mu));
    float is = rsqrtf(var + 1e-5f);
    z0[0][t] = g2[t] * (f0 - mu) * is + be2[t];
    z0[1][t] = g2[t] * (f1 - mu) * is + be2[t];
  }
  __syncthreads();
  if (t < 32) {
    for (int b = 0; b < 2; ++b) {
      float s = lb2[t];
      for (int k = 0; k < 64; ++k) s += z0[b][k] * lw2[t * 64 + k];
      z2[b][t] = fmaxf(s, 0.f);
    }
  }
  __syncthreads();
  if (t < 32) {
    float f0 = z2[0][t], f1 = z2[1][t];
    float mu = 0.5f * (f0 + f1);
    float var = 0.5f * ((f0 - mu) * (f0 - mu) + (f1 - mu) * (f1 - mu));
    float is = rsqrtf(var + 1e-5f);
    z2[0][t] = g3[t] * (f0 - mu) * is + be3[t];
    z2[1][t] = g3[t] * (f1 - mu) * is + be3[t];
  }
  __syncthreads();
  if (t < 10) {
    for (int b = 0; b < 2; ++b) {
      float s = lb3[t];
      for (int k = 0; k < 32; ++k) s += z2[b][k] * lw3[t * 32 + k];
      out[b * 10 + t] = s;
    }
  }
}

// ---------------------------------------------------------------------------
// Host orchestration
// ---------------------------------------------------------------------------

struct LCfg { int kind, bi, bo, ci, co, P; double gbeta; };
static const LCfg CFG[14] = {
  {0, 30, 30,  1,  8,  6, M_PI / 16},
  {1, 30, 15,  8, 16, 36, M_PI / 16},
  {1, 15, 15, 16, 16, 36, M_PI / 8},
  {1, 15,  7, 16, 24, 36, M_PI / 8},
  {1,  7,  7, 24, 24, 36, M_PI / 4},
  {1,  7,  3, 24, 32, 36, M_PI / 4},
  {1,  3,  3, 32, 64, 36, M_PI / 2},
  {1,  3,  3, 64, 32, 36, M_PI / 2},
  {1,  3,  7, 32, 24, 36, M_PI / 4},
  {1,  7,  7, 24, 24, 36, M_PI / 4},
  {1,  7, 15, 24, 16, 36, M_PI / 8},
  {1, 15, 15, 16, 16, 36, M_PI / 8},
  {1, 15, 30, 16,  8, 36, M_PI / 16},
  {1, 30, 30,  8, 64, 36, M_PI / 16},
};

static inline dim3 g1d(long n) {
  long b = (n + 255) / 256;
  if (b < 1) b = 1;
  if (b > 1048576) b = 1048576;
  return dim3((unsigned)b);
}

extern "C" void kernel_launch(void* const* d_in, const int* in_sizes, int n_in,
                              void* d_out, int out_size, void* d_ws, size_t ws_size,
                              hipStream_t stream) {
  (void)in_sizes; (void)n_in; (void)out_size; (void)ws_size;
  const float* x = (const float*)d_in[0];
  const float* cw[14]; const float* cbv[14];
  for (int i = 0; i < 14; ++i) {
    cw[i]  = (const float*)d_in[1 + 2 * i];
    cbv[i] = (const float*)d_in[2 + 2 * i];
  }
  const float* g1  = (const float*)d_in[29];
  const float* be1 = (const float*)d_in[30];
  const float* lw1 = (const float*)d_in[31];
  const float* lb1 = (const float*)d_in[32];
  const float* g2  = (const float*)d_in[33];
  const float* be2 = (const float*)d_in[34];
  const float* lw2 = (const float*)d_in[35];
  const float* lb2 = (const float*)d_in[36];
  const float* g3  = (const float*)d_in[37];
  const float* be3 = (const float*)d_in[38];
  const float* lw3 = (const float*)d_in[39];
  const float* lb3 = (const float*)d_in[40];
  float* out = (float*)d_out;

  char* base = (char*)d_ws;
  size_t off = 0;
  auto alloc = [&](long nfl) -> float* {
    float* p = (float*)(base + off);
    off += ((size_t)nfl * 4 + 255) & ~(size_t)255;
    return p;
  };

  // Persistent buffers: activation ping-pong + tiny tables.
  long maxh = 0;
  for (int i = 0; i < 14; ++i) {
    long s = 2L * CFG[i].co * (2 * CFG[i].bo) * (2 * CFG[i].bo) * (2 * CFG[i].bo);
    if (s > maxh) maxh = s;
  }
  float* hA = alloc(maxh);
  float* hB = alloc(maxh);
  double* lgf = (double*)alloc(256);
  float* w30  = alloc(64);
  float* feat = alloc(128);
  size_t lbase = off;

  k_lgfact<<<dim3(1), 128, 0, stream>>>(lgf);
  k_quadw<<<dim3(1), 64, 0, stream>>>(w30, 30);

  const float* cur = x;
  float* nxt = hA;

  for (int li = 0; li < 14; ++li) {
    off = lbase;                      // per-layer bump-allocator reset
    const LCfg c = CFG[li];
    const int nl = imin_(c.bi, c.bo), M = 2 * nl - 1;
    const int K2 = 2 * c.bi, K2o = 2 * c.bo;
    const long SP2 = (long)M * M;

    float* wq = alloc(K2);
    k_quadw<<<dim3(1), 64, 0, stream>>>(wq, c.bi);

    float* Di = alloc((long)nl * SP2 * K2o);
    k_tab_so3<<<g1d((long)nl * SP2 * K2o), 256, 0, stream>>>(Di, nullptr, c.bo, nl, 0, lgf);

    float* twfr = alloc((long)M * K2),  *twfi = alloc((long)M * K2);
    float* twir = alloc((long)M * K2o), *twii = alloc((long)M * K2o);
    k_twiddle<<<g1d((long)M * K2),  256, 0, stream>>>(twfr, twfi, M, K2,  nl, -1.0);
    k_twiddle<<<g1d((long)M * K2o), 256, 0, stream>>>(twir, twii, M, K2o, nl,  1.0);

    long nyh = 2L * c.co * nl * SP2;
    float* yhr = alloc(nyh), *yhi = alloc(nyh);

    const int Rtot = 2 * M, Stot = c.co * M;
    dim3 gg((Stot + 31) / 32, (Rtot + 31) / 32, nl);

    if (c.kind == 0) {
      // ------- S2 layer (layer 0): 1-D DFT, K-extent = 1 in the GEMM -------
      float* Df = alloc((long)nl * M * K2);
      k_tab_s2df<<<g1d((long)nl * M * K2), 256, 0, stream>>>(Df, wq, c.bi, nl, lgf);
      float* Kbr = alloc((long)nl * M * 6), *Kbi = alloc((long)nl * M * 6);
      k_kb_s2<<<g1d((long)nl * M * 6), 256, 0, stream>>>(Kbr, Kbi, nl, c.gbeta, lgf);

      long leadX = 2L * c.ci * K2;             // (b, c, beta)
      long nXs = leadX * M;
      float* Xsr = alloc(nXs), *Xsi = alloc(nXs);
      k_dft_real<<<g1d(nXs), 256, 0, stream>>>(Xsr, Xsi, cur, twfr, twfi,
                                               leadX, K2, M, 1.f / K2);

      long nxh = 2L * c.ci * nl * M;
      float* xhr = alloc(nxh), *xhi = alloc(nxh);
      k_xh<<<g1d(nxh), 256, 0, stream>>>(xhr, xhi, Xsr, Xsi, Df,
                                         2L * c.ci, nl, (long)M, K2);

      long nps = (long)c.co * c.ci * nl * M;
      float* psr = alloc(nps), *psii = alloc(nps);
      k_psi<<<g1d(nps), 256, 0, stream>>>(psr, psii, cw[li], Kbr, Kbi,
                                          (long)c.co * c.ci, (long)nl * M, 6);

      k_cgemm_wmma<<<gg, 32, 0, stream>>>(xhr, xhi, psr, psii, yhr, yhi,
                                          2, c.ci, c.co, nl, M, 1);
    } else {
      // ------- SO3 layer: 2-D DFT, K-extent = M in the GEMM -------
      float* Df = alloc((long)nl * SP2 * K2);
      k_tab_so3<<<g1d((long)nl * SP2 * K2), 256, 0, stream>>>(Df, wq, c.bi, nl, 1, lgf);
      float* Kbr = alloc((long)nl * SP2 * 36), *Kbi = alloc((long)nl * SP2 * 36);
      k_kb_so3<<<g1d((long)nl * SP2 * 36), 256, 0, stream>>>(Kbr, Kbi, nl, c.gbeta, lgf);

      long lead1 = 2L * c.ci * K2 * K2;        // (b, c, beta, alpha)
      long nT = lead1 * M;
      float* Tr = alloc(nT), *Ti = alloc(nT);
      k_dft_real<<<g1d(nT), 256, 0, stream>>>(Tr, Ti, cur, twfr, twfi,
                                              lead1, K2, M, 1.f / K2);

      long nXs = 2L * c.ci * K2 * SP2;
      float* Xsr = alloc(nXs), *Xsi = alloc(nXs);
      k_dft_mid<<<g1d(nXs), 256, 0, stream>>>(Xsr, Xsi, Tr, Ti, twfr, twfi,
                                              2L * c.ci * K2, K2, M, M, 1.f / K2);

      long nxh = 2L * c.ci * nl * SP2;
      float* xhr = alloc(nxh), *xhi = alloc(nxh);
      k_xh<<<g1d(nxh), 256, 0, stream>>>(xhr, xhi, Xsr, Xsi, Df,
                                         2L * c.ci, nl, SP2, K2);

      long nps = (long)c.co * c.ci * nl * SP2;
      float* psr = alloc(nps), *psii = alloc(nps);
      k_psi<<<g1d(nps), 256, 0, stream>>>(psr, psii, cw[li], Kbr, Kbi,
                                          (long)c.co * c.ci, (long)nl * SP2, 36);

      k_cgemm_wmma<<<gg, 32, 0, stream>>>(xhr, xhi, psr, psii, yhr, yhi,
                                          2, c.ci, c.co, nl, M, M);
    }

    // ------- shared inverse path: Wigner synthesis + inverse DFT2 + bias/ReLU
    long nYs = 2L * c.co * K2o * SP2;
    float* Ysr = alloc(nYs), *Ysi = alloc(nYs);
    k_ys<<<g1d(nYs), 256, 0, stream>>>(Ysr, Ysi, yhr, yhi, Di,
                                       2L * c.co, nl, SP2, K2o);

    long nU = 2L * c.co * K2o * M * K2o;
    float* Ur = alloc(nU), *Ui = alloc(nU);
    k_idft_last<<<g1d(nU), 256, 0, stream>>>(Ur, Ui, Ysr, Ysi, twir, twii,
                                             2L * c.co * K2o * M, M, K2o);

    long ny = 2L * c.co * K2o * K2o * K2o;
    k_idft_mid_bias_relu<<<g1d(ny), 256, 0, stream>>>(nxt, Ur, Ui, twir, twii,
                                                      cbv[li], 2L * c.co * K2o,
                                                      M, K2o, c.co);

    cur = nxt;
    nxt = (nxt == hA) ? hB : hA;
  }

  // Head: quadrature pooling over (beta,alpha,gamma) then BN/MLP.
  k_feat<<<dim3(2 * 64), 256, 0, stream>>>(feat, cur, w30, 60);
  k_head<<<dim3(1), 64, 0, stream>>>(feat, g1, be1, lw1, lb1,
                                     g2, be2, lw2, lb2, g3, be3, lw3, lb3, out);
}